// NodeGCN_17583596110090
// MI455X (gfx1250) — compile-verified
//
#include <hip/hip_runtime.h>
#include <hip/hip_bf16.h>

#define IN_DIM 512
#define HID 20
#define OUT_DIM 10

typedef __attribute__((ext_vector_type(2))) float v2f;
typedef __attribute__((ext_vector_type(8))) float v8f;

// ---- init: zero emb concat buffer, deg = 1.0 (self-loop weight) ----
__global__ void k_init(float* __restrict__ deg, float* __restrict__ emb, int n) {
  long long t = (long long)blockIdx.x * blockDim.x + threadIdx.x;
  if (t < (long long)n * 64) emb[t] = 0.0f;
  if (t < n) deg[t] = 1.0f;
}

// ---- pack weight matrix W[Krows x Ncols] into per-lane WMMA B-fragment order ----
// Bp[((ks*ntiles + t)*32 + lane)*2 + j] = W[4*ks + 2*(lane>>4) + j][t*16 + (lane&15)]
// so each lane's B fragment for one K-step is a single aligned b64 load.
__global__ void k_packB(const float* __restrict__ W, int Krows, int Ncols,
                        float* __restrict__ Bp, int Ksteps, int ntiles) {
  int t = blockIdx.x * blockDim.x + threadIdx.x;
  int total = Ksteps * ntiles * 64;
  if (t >= total) return;
  int j = t & 1;
  int lane = (t >> 1) & 31;
  int tile = (t >> 6) % ntiles;
  int ks = (t >> 6) / ntiles;
  int k = 4 * ks + 2 * (lane >> 4) + j;
  int n = tile * 16 + (lane & 15);
  Bp[t] = (k < Krows && n < Ncols) ? W[k * Ncols + n] : 0.0f;
}

// ---- in-degree accumulation over edges ----
__global__ void k_deg(const int* __restrict__ dst, const float* __restrict__ ew,
                      float* __restrict__ deg, long long E) {
  long long e = (long long)blockIdx.x * blockDim.x + threadIdx.x;
  if (e >= E) return;
  unsafeAtomicAdd(&deg[dst[e]], ew[e]);
}

__global__ void k_dinv(const float* __restrict__ deg, float* __restrict__ dinv, int n) {
  int i = blockIdx.x * blockDim.x + threadIdx.x;
  if (i >= n) return;
  float d = deg[i];
  dinv[i] = (d > 0.0f) ? rsqrtf(d) : 0.0f;
}

// ---- self-loop term: agg[i] = h[i] * dinv[i]^2 ----
__global__ void k_selfloop(const float* __restrict__ h, const float* __restrict__ dinv,
                           float* __restrict__ agg, int n) {
  int t = blockIdx.x * blockDim.x + threadIdx.x;
  int i = t >> 5, c = t & 31;
  if (i >= n) return;
  if (c < HID) { float di = dinv[i]; agg[(size_t)i * 32 + c] = h[(size_t)i * 32 + c] * di * di; }
}

// ---- edge scatter: agg[dst] += h[src] * dinv[src]*ew*dinv[dst]  (L2-resident) ----
__global__ void k_edge_agg(const int* __restrict__ src, const int* __restrict__ dst,
                           const float* __restrict__ ew, const float* __restrict__ dinv,
                           const float* __restrict__ h, float* __restrict__ agg, long long E) {
  long long t = (long long)blockIdx.x * blockDim.x + threadIdx.x;
  long long e = t >> 5;
  int c = (int)(t & 31);
  if (e >= E) return;
  int s = src[e], d = dst[e];
  float coef = dinv[s] * ew[e] * dinv[d];
  if (c < HID) unsafeAtomicAdd(&agg[(size_t)d * 32 + c], h[(size_t)s * 32 + c] * coef);
}

// ---- + bias, row L2-normalize (eps 1e-12), ReLU, write into emb concat buffer ----
__global__ void k_bias_norm(const float* __restrict__ agg, const float* __restrict__ b,
                            float* __restrict__ emb, int n, int off) {
  int t = blockIdx.x * blockDim.x + threadIdx.x;
  int i = t >> 5, c = t & 31;
  if (i >= n) return;
  float v = 0.0f;
  if (c < HID) v = agg[(size_t)i * 32 + c] + b[c];
  float s = v * v;
#pragma unroll
  for (int m = 16; m >= 1; m >>= 1) s += __shfl_xor(s, m, 32);
  float nrm = fmaxf(sqrtf(s), 1e-12f);
  if (c < HID) {
    float r = v / nrm;
    emb[(size_t)i * 64 + off + c] = (r > 0.0f) ? r : 0.0f;
  }
}

// ---- fp32 WMMA GEMM, M=32 per wave (two 16-row tiles sharing B fragments) ----
// A fragment (16x4 f32): lanes 0-15 hold K={k0,k0+1}, lanes 16-31 hold K={k0+2,k0+3}.
// B comes pre-packed in fragment order (see k_packB): one b64 load per fragment.
// C/D (16x16): VGPR r = rows {r, r+8} across lane halves, N = lane&15.
template <int NTILES>
__global__ void k_wmma_gemm(const float* __restrict__ A, int lda, int acol0,
                            const float* __restrict__ Bp, int Ksteps,
                            float* __restrict__ out, int ldo, int ncols,
                            const float* __restrict__ bias, int npairs) {
  int pair = blockIdx.x * (blockDim.x >> 5) + (threadIdx.x >> 5);
  if (pair >= npairs) return;  // wave-uniform: EXEC stays all-1s for WMMA
  int lane = threadIdx.x & 31;
  int half = lane >> 4;
  int l16 = lane & 15;
  int m0 = pair * 32;
  const float* ap0 = A + (size_t)(m0 + l16) * lda + acol0 + half * 2;
  const float* ap1 = ap0 + (size_t)16 * lda;
  const float* bp = Bp + lane * 2;
  v8f acc[2][NTILES];
#pragma unroll
  for (int rr = 0; rr < 2; ++rr)
#pragma unroll
    for (int t = 0; t < NTILES; ++t) acc[rr][t] = (v8f){0.f, 0.f, 0.f, 0.f, 0.f, 0.f, 0.f, 0.f};

#pragma unroll 2
  for (int ks = 0; ks < Ksteps; ++ks) {
    v2f a0 = *(const v2f*)(ap0 + 4 * ks);
    v2f a1 = *(const v2f*)(ap1 + 4 * ks);
#pragma unroll
    for (int t = 0; t < NTILES; ++t) {
      v2f b = *(const v2f*)(bp + (size_t)(ks * NTILES + t) * 64);
      acc[0][t] = __builtin_amdgcn_wmma_f32_16x16x4_f32(false, a0, false, b, (short)0, acc[0][t], false, false);
      acc[1][t] = __builtin_amdgcn_wmma_f32_16x16x4_f32(false, a1, false, b, (short)0, acc[1][t], false, false);
    }
  }

#pragma unroll
  for (int rr = 0; rr < 2; ++rr)
#pragma unroll
    for (int t = 0; t < NTILES; ++t)
#pragma unroll
      for (int r = 0; r < 8; ++r) {
        int row = m0 + rr * 16 + r + half * 8;
        int col = t * 16 + l16;
        if (col < ncols) out[(size_t)row * ldo + col] = acc[rr][t][r] + (bias ? bias[col] : 0.0f);
      }
}

extern "C" void kernel_launch(void* const* d_in, const int* in_sizes, int n_in,
                              void* d_out, int out_size, void* d_ws, size_t ws_size,
                              hipStream_t stream) {
  const float* x  = (const float*)d_in[0];
  const int*   ei = (const int*)d_in[1];
  const float* ew = (const float*)d_in[2];
  const float* W1 = (const float*)d_in[3];
  const float* b1 = (const float*)d_in[4];
  const float* W2 = (const float*)d_in[5];
  const float* b2 = (const float*)d_in[6];
  const float* W3 = (const float*)d_in[7];
  const float* b3 = (const float*)d_in[8];
  const float* Wl = (const float*)d_in[9];
  const float* bl = (const float*)d_in[10];
  float* out = (float*)d_out;

  const int N = in_sizes[0] / IN_DIM;  // 100000
  const long long E = in_sizes[2];     // 3.2M
  const int* src = ei;
  const int* dst = ei + E;

  // K-steps (K rounded up to multiple of 4) per GEMM
  const int KS1 = IN_DIM / 4;       // 128
  const int KS2 = (HID + 3) / 4;    // 5
  const int KSL = 64 / 4;           // 16

  float* ws = (float*)d_ws;
  float* deg  = ws; ws += N;
  float* dinv = ws; ws += N;
  float* h    = ws; ws += (size_t)N * 32;
  float* agg  = ws; ws += (size_t)N * 32;
  float* emb  = ws; ws += (size_t)N * 64;
  float* W1p  = ws; ws += KS1 * 2 * 64;  // 16384
  float* W2p  = ws; ws += KS2 * 2 * 64;  // 640
  float* W3p  = ws; ws += KS2 * 2 * 64;  // 640
  float* Wlp  = ws; ws += KSL * 1 * 64;  // 1024

  const int TB = 256;
  const int NPAIRS = N / 32;            // 3125 (two 16-row tiles per wave)
  dim3 gblk((NPAIRS + 3) / 4);          // 4 waves/block
  long long nt32 = (long long)N * 32;
  long long et32 = E * 32;
  int g_n32 = (int)((nt32 + TB - 1) / TB);
  int g_e32 = (int)((et32 + TB - 1) / TB);

  k_init<<<(int)(((long long)N * 64 + TB - 1) / TB), TB, 0, stream>>>(deg, emb, N);
  k_packB<<<(KS1 * 2 * 64 + TB - 1) / TB, TB, 0, stream>>>(W1, IN_DIM, HID, W1p, KS1, 2);
  k_packB<<<(KS2 * 2 * 64 + TB - 1) / TB, TB, 0, stream>>>(W2, HID, HID, W2p, KS2, 2);
  k_packB<<<(KS2 * 2 * 64 + TB - 1) / TB, TB, 0, stream>>>(W3, HID, HID, W3p, KS2, 2);
  k_packB<<<(KSL * 1 * 64 + TB - 1) / TB, TB, 0, stream>>>(Wl, 3 * HID, OUT_DIM, Wlp, KSL, 1);
  k_deg<<<(int)((E + TB - 1) / TB), TB, 0, stream>>>(dst, ew, deg, E);
  k_dinv<<<(N + TB - 1) / TB, TB, 0, stream>>>(deg, dinv, N);

  // ---- layer 1: h = x @ W1 ----
  k_wmma_gemm<2><<<gblk, 128, 0, stream>>>(x, IN_DIM, 0, W1p, KS1, h, 32, 32, nullptr, NPAIRS);
  k_selfloop<<<g_n32, TB, 0, stream>>>(h, dinv, agg, N);
  k_edge_agg<<<g_e32, TB, 0, stream>>>(src, dst, ew, dinv, h, agg, E);
  k_bias_norm<<<g_n32, TB, 0, stream>>>(agg, b1, emb, N, 0);

  // ---- layer 2: h = o1 @ W2 ----  (K padded 20->ok: B rows >=20 are zero,
  // A cols 0..19 only; step 5 reads A col 16..19 fine since 4*KS2 = 20 exactly)
  k_wmma_gemm<2><<<gblk, 128, 0, stream>>>(emb, 64, 0, W2p, KS2, h, 32, 32, nullptr, NPAIRS);
  k_selfloop<<<g_n32, TB, 0, stream>>>(h, dinv, agg, N);
  k_edge_agg<<<g_e32, TB, 0, stream>>>(src, dst, ew, dinv, h, agg, E);
  k_bias_norm<<<g_n32, TB, 0, stream>>>(agg, b2, emb, N, HID);

  // ---- layer 3: h = o2 @ W3 ----
  k_wmma_gemm<2><<<gblk, 128, 0, stream>>>(emb, 64, HID, W3p, KS2, h, 32, 32, nullptr, NPAIRS);
  k_selfloop<<<g_n32, TB, 0, stream>>>(h, dinv, agg, N);
  k_edge_agg<<<g_e32, TB, 0, stream>>>(src, dst, ew, dinv, h, agg, E);
  k_bias_norm<<<g_n32, TB, 0, stream>>>(agg, b3, emb, N, 2 * HID);

  // ---- final linear: out = emb[:, 0:60 (zero-padded to 64)] @ Wl + bl ----
  k_wmma_gemm<1><<<gblk, 128, 0, stream>>>(emb, 64, 0, Wlp, KSL, out, OUT_DIM, OUT_DIM, bl, NPAIRS);
}